// KnittedVQC_33844342292918
// MI455X (gfx1250) — compile-verified
//
#include <hip/hip_runtime.h>

// ---------------------------------------------------------------------------
// KnittedVQC on MI455X (gfx1250, wave32).
// One wave = one 8-qubit circuit. State = 16x16 complex f32 matrix.
// Per layer: RY-row (real 16x16 A), Rot-row (complex A), RY-col (on S^T),
// Rot-col (on S^T) -- each a chain of V_WMMA_F32_16X16X4_F32 -- then the
// CNOT ring as a prefix-XOR permutation folded into the LDS scatter.
// ---------------------------------------------------------------------------

typedef float v2f __attribute__((ext_vector_type(2)));
typedef float v8f __attribute__((ext_vector_type(8)));

#define N_SUBQ   4
#define N_LAYERS 3
#define WAVES    8   // 256-thread block = 8 wave32

__device__ __forceinline__ v8f wmma4(v2f a, v2f b, v8f c) {
#if __has_builtin(__builtin_amdgcn_wmma_f32_16x16x4_f32)
  // 8-arg pattern: (neg_a, A, neg_b, B, c_mod, C, reuse_a, reuse_b)
  return __builtin_amdgcn_wmma_f32_16x16x4_f32(false, a, false, b, (short)0, c,
                                               false, false);
#else
  v8f d = c;
  asm volatile("v_wmma_f32_16x16x4_f32 %0, %1, %2, %0"
               : "+v"(d) : "v"(a), "v"(b));
  return d;
#endif
}

__device__ __forceinline__ float2 cmulf2(float2 a, float2 b) {
  return make_float2(a.x * b.x - a.y * b.y, a.x * b.y + a.y * b.x);
}

// One matmul stage: D = A * B, where B is the 16x16 complex state read from
// wave-private LDS (optionally transposed), result written back (optionally
// through the CNOT-ring permutation).
template <bool TRANS_READ, bool CNOT_STORE, bool COMPLEX_A>
__device__ __forceinline__ void apply_stage(float2* __restrict__ st,
                                            const v2f* Ar, const v2f* Ai,
                                            int laneM, int hh) {
  v8f Dr = {};
  v8f Di = {};
#pragma unroll
  for (int c = 0; c < 4; ++c) {
    const int K0 = 4 * c + 2 * hh;  // K for v=0; v=1 is K0+1
    float2 b0, b1;
    if (TRANS_READ) {  // B[K][N] = S[N][K]
      b0 = st[(laneM << 4) + K0];
      b1 = st[(laneM << 4) + K0 + 1];
    } else {           // B[K][N] = S[K][N]
      b0 = st[(K0 << 4) + laneM];
      b1 = st[((K0 + 1) << 4) + laneM];
    }
    v2f Br = {b0.x, b1.x};
    v2f Bi = {b0.y, b1.y};
    Dr = wmma4(Ar[c], Br, Dr);
    Di = wmma4(Ar[c], Bi, Di);
    if constexpr (COMPLEX_A) {
      // f32 WMMA has no A/B NEG modifier (ISA VOP3P table) -> negate B copy.
      v2f nBi = {-b0.y, -b1.y};
      Dr = wmma4(Ai[c], nBi, Dr);  // Dr += (-Ai)*Bi
      Di = wmma4(Ai[c], Br, Di);   // Di +=   Ai *Br
    }
  }
#pragma unroll
  for (int r = 0; r < 8; ++r) {
    float2 v = make_float2(Dr[r], Di[r]);
    if constexpr (CNOT_STORE) {
      // Regs hold S^T: element (M=r+8h, N=laneM) is state row N, col M.
      int i = (laneM << 4) + r + 8 * hh;          // basis index, wire0 = MSB
      int t = i ^ (i >> 1); t ^= (t >> 2); t ^= (t >> 4);  // suffix-XOR bits
      int ip = (t & 0x7F) | (((t ^ (i >> 7)) & 1) << 7);   // ring-wrap bit
      st[ip] = v;                                  // CNOT ring = permutation
    } else {
      st[((r + 8 * hh) << 4) + laneM] = v;         // plain C/D layout store
    }
  }
}

__global__ __launch_bounds__(256) void vqc_kernel(
    const float* __restrict__ x, const float* __restrict__ w,
    const float* __restrict__ scale, float* __restrict__ out, int B) {
  __shared__ float2 s_gate[N_LAYERS * 8 * 4];   // 2x2 Rot gates
  __shared__ float2 s_rot[N_LAYERS * 2 * 256];  // 16x16 tensor-product Rots
  __shared__ float2 s_st[WAVES][256];           // per-wave state tiles

  const int sub = blockIdx.y;
  const int tid = threadIdx.x;

  // ---- per-block setup: 2x2 Rot(phi,theta,omega) gate entries -------------
  if (tid < N_LAYERS * 8 * 4) {
    int g = tid >> 2, e = tid & 3;
    int layer = g >> 3, q = g & 7;
    const float* wp = w + ((sub * N_LAYERS + layer) * 8 + q) * 3;
    float phi = wp[0], th = wp[1], om = wp[2];
    float ch = __cosf(0.5f * th), sh = __sinf(0.5f * th);
    int m = e >> 1, k = e & 1;
    float ang, mag;
    if (m == 0 && k == 0)      { ang = -0.5f * (phi + om); mag =  ch; }
    else if (m == 0 && k == 1) { ang =  0.5f * (phi - om); mag = -sh; }
    else if (m == 1 && k == 0) { ang = -0.5f * (phi - om); mag =  sh; }
    else                       { ang =  0.5f * (phi + om); mag =  ch; }
    s_gate[tid] = make_float2(mag * __cosf(ang), mag * __sinf(ang));
  }
  __syncthreads();

  // ---- per-block setup: 4-qubit tensor products -> 16x16 complex ----------
  for (int idx = tid; idx < N_LAYERS * 2 * 256; idx += 256) {
    int layer = idx >> 9;
    int side = (idx >> 8) & 1;           // 0 = row qubits 0-3, 1 = col 4-7
    int M = (idx >> 4) & 15, K = idx & 15;
    float2 acc = make_float2(1.f, 0.f);
#pragma unroll
    for (int j = 0; j < 4; ++j) {
      int mb = (M >> (3 - j)) & 1, kb = (K >> (3 - j)) & 1;
      float2 g = s_gate[((layer * 8 + side * 4 + j) << 2) + (mb << 1) + kb];
      acc = cmulf2(acc, g);
    }
    s_rot[idx] = acc;
  }
  __syncthreads();

  const int lane = tid & 31, wv = tid >> 5;
  const int laneM = lane & 15, hh = lane >> 4;
  float2* st = s_st[wv];

  // ---- Rot A-operands into registers: [layer*2+side][chunk], A[M][K] ------
  v2f rotAr[N_LAYERS * 2 * 4], rotAi[N_LAYERS * 2 * 4];
#pragma unroll
  for (int ls = 0; ls < N_LAYERS * 2; ++ls) {
#pragma unroll
    for (int c = 0; c < 4; ++c) {
      float2 e0 = s_rot[(ls << 8) + (laneM << 4) + (4 * c + 2 * hh)];
      float2 e1 = s_rot[(ls << 8) + (laneM << 4) + (4 * c + 1 + 2 * hh)];
      rotAr[ls * 4 + c] = (v2f){e0.x, e1.x};
      rotAi[ls * 4 + c] = (v2f){e0.y, e1.y};
    }
  }

  const int waveGlobal = blockIdx.x * WAVES + wv;
  const int nWaves = gridDim.x * WAVES;

#pragma unroll 1
  for (int s = waveGlobal; s < B; s += nWaves) {
    // Embedding angles (wave-uniform address -> scalar loads)
    const float* xs = x + (size_t)s * (N_SUBQ * 8) + sub * 8;
    float cq[8], sq[8];
#pragma unroll
    for (int q = 0; q < 8; ++q) {
      float a = 0.5f * xs[q];
      cq[q] = __cosf(a);
      sq[q] = __sinf(a);
    }
    // RY tensor-product A-operands (real), [side][chunk]; same every layer.
    v2f ryA[8];
#pragma unroll
    for (int side = 0; side < 2; ++side) {
#pragma unroll
      for (int c = 0; c < 4; ++c) {
        float pv[2];
#pragma unroll
        for (int v = 0; v < 2; ++v) {
          int K = 4 * c + v + 2 * hh;
          float p = 1.f;
#pragma unroll
          for (int j = 0; j < 4; ++j) {
            int mb = (laneM >> (3 - j)) & 1, kb = (K >> (3 - j)) & 1;
            int q = side * 4 + j;
            p *= (mb == kb) ? cq[q] : (mb ? sq[q] : -sq[q]);
          }
          pv[v] = p;
        }
        ryA[side * 4 + c] = (v2f){pv[0], pv[1]};
      }
    }

    // |0...0>
#pragma unroll
    for (int k = 0; k < 8; ++k) st[lane + 32 * k] = make_float2(0.f, 0.f);
    if (lane == 0) st[0] = make_float2(1.f, 0.f);

#pragma unroll
    for (int layer = 0; layer < N_LAYERS; ++layer) {
      // RY on row qubits:  S <- A_ry_row * S
      apply_stage<false, false, false>(st, &ryA[0], nullptr, laneM, hh);
      // Rot on row qubits: S <- A_rot_row * S
      apply_stage<false, false, true>(st, &rotAr[(2 * layer + 0) * 4],
                                      &rotAi[(2 * layer + 0) * 4], laneM, hh);
      // RY on col qubits:  S^T <- A_ry_col * S^T  (transposed read)
      apply_stage<true, false, false>(st, &ryA[4], nullptr, laneM, hh);
      // Rot on col qubits + CNOT-ring permutation on store
      apply_stage<false, true, true>(st, &rotAr[(2 * layer + 1) * 4],
                                     &rotAi[(2 * layer + 1) * 4], laneM, hh);
    }

    // ---- <Z_w> = sum_i sign_w(i) * |amp_i|^2 ------------------------------
    float acc[8] = {0.f, 0.f, 0.f, 0.f, 0.f, 0.f, 0.f, 0.f};
#pragma unroll
    for (int r = 0; r < 8; ++r) {
      int i = lane + 32 * r;
      float2 a = st[i];
      float p = a.x * a.x + a.y * a.y;
#pragma unroll
      for (int wq = 0; wq < 8; ++wq)
        acc[wq] += ((i >> (7 - wq)) & 1) ? -p : p;
    }
#pragma unroll
    for (int off = 16; off > 0; off >>= 1) {
#pragma unroll
      for (int wq = 0; wq < 8; ++wq)
        acc[wq] += __shfl_xor(acc[wq], off, 32);
    }
    if (lane == 0) {
#pragma unroll
      for (int wq = 0; wq < 8; ++wq)
        out[(size_t)s * (N_SUBQ * 8) + sub * 8 + wq] =
            acc[wq] * scale[sub * 8 + wq];
    }
  }
}

extern "C" void kernel_launch(void* const* d_in, const int* in_sizes, int n_in,
                              void* d_out, int out_size, void* d_ws,
                              size_t ws_size, hipStream_t stream) {
  const float* x = (const float*)d_in[0];      // (B, 32) f32
  const float* w = (const float*)d_in[1];      // (4, 3, 8, 3) f32
  const float* scale = (const float*)d_in[2];  // (32,) f32
  float* out = (float*)d_out;                  // (B, 32) f32
  int B = in_sizes[0] / (N_SUBQ * 8);

  dim3 grid(1024, N_SUBQ, 1);  // 8192 waves per sub-circuit -> 4 samples/wave
  vqc_kernel<<<grid, 256, 0, stream>>>(x, w, scale, out, B);
}